// GIN_60009283060273
// MI455X (gfx1250) — compile-verified
//
#include <hip/hip_runtime.h>
#include <hip/hip_bf16.h>

// GIN 3-layer forward for MI455X (gfx1250, wave32).
// Memory-bound workload (~1.4 GB traffic, ~5.5 GFLOP): edge segment-sums use
// native global_atomic_add_f32; per-node MLPs use V_WMMA_F32_16X16X4_F32
// (exact fp32) with one wave computing a 16-node x 96-channel tile.

typedef __attribute__((ext_vector_type(2))) float v2f;
typedef __attribute__((ext_vector_type(8))) float v8f;

#define N_NODES 50000
#define N_EDGES 800000
#define N_GRAPH 256
#define F_IN    32
#define F_H     96
#define BN_EPS  1e-5f
#define ZSTRIDE 100   // 96 + 4 pad, keeps float2 LDS reads 8B aligned

// ---------------------------------------------------------------------------
// Edge aggregation: agg[dst] += h[src]  (float4 per thread, native f32 atomics)
// ---------------------------------------------------------------------------
template <int F>
__global__ void aggregate_kernel(const float* __restrict__ h,
                                 const int*   __restrict__ edge_index,
                                 float*       __restrict__ agg) {
    const int work = N_EDGES * (F / 4);
    int tid = blockIdx.x * blockDim.x + threadIdx.x;
    if (tid >= work) return;
    const int e  = tid / (F / 4);
    const int c4 = (tid % (F / 4)) * 4;
    const int src = edge_index[e];             // row 0 of edge_index
    const int dst = edge_index[N_EDGES + e];   // row 1 of edge_index
    const float4 v = *(const float4*)&h[(size_t)src * F + c4];
    float* out = &agg[(size_t)dst * F + c4];
    unsafeAtomicAdd(out + 0, v.x);
    unsafeAtomicAdd(out + 1, v.y);
    unsafeAtomicAdd(out + 2, v.z);
    unsafeAtomicAdd(out + 3, v.w);
}

// ---------------------------------------------------------------------------
// Fused GIN MLP for a 16-node tile, one wave per block.
//   z  = relu(BN((hin+agg) @ w_in + b_in)) ; hout = relu(z @ w_out + b_out)
// A-frag (16x4 f32): lanes 0-15 rows M=0..15 hold K={k0,k0+1}; lanes 16-31
// hold K={k0+2,k0+3}. C/D (16x16 f32): VGPR r = row (r | half*8), lane%16 = col.
// ---------------------------------------------------------------------------
template <int K>
__global__ __launch_bounds__(32)
void gin_mlp_kernel(const float* __restrict__ hin,   // [N, K]
                    const float* __restrict__ agg,   // [N, K]
                    const float* __restrict__ w_in,  // [K, 96]
                    const float* __restrict__ b_in,
                    const float* __restrict__ gamma,
                    const float* __restrict__ beta,
                    const float* __restrict__ rmean,
                    const float* __restrict__ rvar,
                    const float* __restrict__ w_out, // [96, 96]
                    const float* __restrict__ b_out,
                    float*       __restrict__ hout)  // [N, 96]
{
    __shared__ float zlds[16 * ZSTRIDE];

    const int lane  = threadIdx.x;        // 0..31
    const int l15   = lane & 15;
    const int koff  = (lane >> 4) * 2;    // 0 or 2 : K offset for this half
    const int mbase = (lane >> 4) * 8;    // 0 or 8 : row offset of C half
    const int nb    = blockIdx.x * 16;    // node tile base (N % 16 == 0)
    const int row   = nb + l15;

    // ---------------- GEMM1: (hin + agg) @ w_in -> acc[6] ----------------
    v8f acc[6] = {};
    for (int k0 = 0; k0 < K; k0 += 4) {
        const int ka = k0 + koff;
        v2f a;
        {
            const float2 xv = *(const float2*)&hin[(size_t)row * K + ka];
            const float2 av = *(const float2*)&agg[(size_t)row * K + ka];
            a.x = xv.x + av.x;
            a.y = xv.y + av.y;
        }
#pragma unroll
        for (int t = 0; t < 6; ++t) {
            v2f b;
            b.x = w_in[(size_t)(ka + 0) * F_H + t * 16 + l15];
            b.y = w_in[(size_t)(ka + 1) * F_H + t * 16 + l15];
            acc[t] = __builtin_amdgcn_wmma_f32_16x16x4_f32(
                false, a, false, b, (short)0, acc[t], false, false);
        }
    }

    // ------------- bias + eval-mode BN + ReLU, stage z into LDS -------------
#pragma unroll
    for (int t = 0; t < 6; ++t) {
        const int c = t * 16 + l15;
        const float sc = gamma[c] * rsqrtf(rvar[c] + BN_EPS);
        const float sh = beta[c] + (b_in[c] - rmean[c]) * sc;
#pragma unroll
        for (int r = 0; r < 8; ++r) {
            float z = acc[t][r] * sc + sh;
            zlds[(mbase + r) * ZSTRIDE + c] = fmaxf(z, 0.0f);
        }
    }
    __syncthreads();

    // ---------------- GEMM2: z @ w_out -> acc2[6] ----------------
    v8f acc2[6] = {};
    for (int k0 = 0; k0 < F_H; k0 += 4) {
        const int ka = k0 + koff;
        v2f a;
        {
            const float2 zv = *(const float2*)&zlds[l15 * ZSTRIDE + ka];
            a.x = zv.x;
            a.y = zv.y;
        }
#pragma unroll
        for (int t = 0; t < 6; ++t) {
            v2f b;
            b.x = w_out[(size_t)(ka + 0) * F_H + t * 16 + l15];
            b.y = w_out[(size_t)(ka + 1) * F_H + t * 16 + l15];
            acc2[t] = __builtin_amdgcn_wmma_f32_16x16x4_f32(
                false, a, false, b, (short)0, acc2[t], false, false);
        }
    }

    // ---------------- bias + ReLU, store ----------------
#pragma unroll
    for (int t = 0; t < 6; ++t) {
        const int c  = t * 16 + l15;
        const float bo = b_out[c];
#pragma unroll
        for (int r = 0; r < 8; ++r) {
            const float z = fmaxf(acc2[t][r] + bo, 0.0f);
            hout[(size_t)(nb + mbase + r) * F_H + c] = z;
        }
    }
}

// ---------------------------------------------------------------------------
// Graph pooling: pooled[batch[n]] += h[n]
// ---------------------------------------------------------------------------
__global__ void pool_kernel(const float* __restrict__ h,
                            const int*   __restrict__ batch,
                            float*       __restrict__ pooled) {
    const int work = N_NODES * (F_H / 4);
    int tid = blockIdx.x * blockDim.x + threadIdx.x;
    if (tid >= work) return;
    const int n  = tid / (F_H / 4);
    const int c4 = (tid % (F_H / 4)) * 4;
    const int g  = batch[n];
    const float4 v = *(const float4*)&h[(size_t)n * F_H + c4];
    float* out = &pooled[(size_t)g * F_H + c4];
    unsafeAtomicAdd(out + 0, v.x);
    unsafeAtomicAdd(out + 1, v.y);
    unsafeAtomicAdd(out + 2, v.z);
    unsafeAtomicAdd(out + 3, v.w);
}

// ---------------------------------------------------------------------------
// Head: out[b] = dot(pooled[b], head_w[r_target[b]]) + head_b[r_target[b]]
// ---------------------------------------------------------------------------
__global__ void head_kernel(const float* __restrict__ pooled,
                            const int*   __restrict__ r_target,
                            const float* __restrict__ head_w,  // [T, 96]
                            const float* __restrict__ head_b,  // [T]
                            float*       __restrict__ out) {
    int b = blockIdx.x * blockDim.x + threadIdx.x;
    if (b >= N_GRAPH) return;
    const int t = r_target[b];
    const float* w = &head_w[(size_t)t * F_H];
    const float* p = &pooled[(size_t)b * F_H];
    float s = 0.0f;
#pragma unroll 8
    for (int c = 0; c < F_H; ++c) s += p[c] * w[c];
    out[b] = s + head_b[t];
}

// ---------------------------------------------------------------------------
extern "C" void kernel_launch(void* const* d_in, const int* in_sizes, int n_in,
                              void* d_out, int out_size, void* d_ws, size_t ws_size,
                              hipStream_t stream) {
    const float* x          = (const float*)d_in[0];
    const int*   edge_index = (const int*)  d_in[1];
    const int*   batch      = (const int*)  d_in[2];
    const int*   r_target   = (const int*)  d_in[3];
    // per-layer params: w_in, b_in, gamma, beta, rmean, rvar, w_out, b_out
    const float* P[3][8];
    for (int i = 0; i < 3; ++i)
        for (int j = 0; j < 8; ++j)
            P[i][j] = (const float*)d_in[4 + i * 8 + j];
    const float* head_w = (const float*)d_in[28];
    const float* head_b = (const float*)d_in[29];
    float* out = (float*)d_out;

    // workspace layout
    char* ws = (char*)d_ws;
    const size_t nodeF = (size_t)N_NODES * F_H * sizeof(float);
    float* agg    = (float*)(ws);
    float* hA     = (float*)(ws + nodeF);
    float* hB     = (float*)(ws + 2 * nodeF);
    float* pooled = (float*)(ws + 3 * nodeF);

    const int TB = 256;

    // ---- layer 1 (K = 32) ----
    hipMemsetAsync(agg, 0, (size_t)N_NODES * F_IN * sizeof(float), stream);
    {
        const int work = N_EDGES * (F_IN / 4);
        aggregate_kernel<F_IN><<<(work + TB - 1) / TB, TB, 0, stream>>>(x, edge_index, agg);
    }
    gin_mlp_kernel<F_IN><<<N_NODES / 16, 32, 0, stream>>>(
        x, agg, P[0][0], P[0][1], P[0][2], P[0][3], P[0][4], P[0][5], P[0][6], P[0][7], hA);

    // ---- layer 2 (K = 96) ----
    hipMemsetAsync(agg, 0, nodeF, stream);
    {
        const int work = N_EDGES * (F_H / 4);
        aggregate_kernel<F_H><<<(work + TB - 1) / TB, TB, 0, stream>>>(hA, edge_index, agg);
    }
    gin_mlp_kernel<F_H><<<N_NODES / 16, 32, 0, stream>>>(
        hA, agg, P[1][0], P[1][1], P[1][2], P[1][3], P[1][4], P[1][5], P[1][6], P[1][7], hB);

    // ---- layer 3 (K = 96) ----
    hipMemsetAsync(agg, 0, nodeF, stream);
    {
        const int work = N_EDGES * (F_H / 4);
        aggregate_kernel<F_H><<<(work + TB - 1) / TB, TB, 0, stream>>>(hB, edge_index, agg);
    }
    gin_mlp_kernel<F_H><<<N_NODES / 16, 32, 0, stream>>>(
        hB, agg, P[2][0], P[2][1], P[2][2], P[2][3], P[2][4], P[2][5], P[2][6], P[2][7], hA);

    // ---- pooling + head ----
    hipMemsetAsync(pooled, 0, (size_t)N_GRAPH * F_H * sizeof(float), stream);
    {
        const int work = N_NODES * (F_H / 4);
        pool_kernel<<<(work + TB - 1) / TB, TB, 0, stream>>>(hA, batch, pooled);
    }
    head_kernel<<<(N_GRAPH + TB - 1) / TB, TB, 0, stream>>>(pooled, r_target, head_w, head_b, out);
}